// get_loss_79207786873276
// MI455X (gfx1250) — compile-verified
//
#include <hip/hip_runtime.h>
#include <hip/hip_bf16.h>
#include <float.h>

typedef float v2f __attribute__((ext_vector_type(2)));
typedef float v8f __attribute__((ext_vector_type(8)));

#define BB   16
#define NPTS 2048
#define NCLS 40
#define HIOFF (2 * NPTS + 32)   // hi-region start in ldsB (32-dword pad: bank-conflict-free)

// One block = one batch b (blockIdx.y) x one 128-row tile of X (blockIdx.x).
// 8 wave32s per block; each wave owns 16 X rows and sweeps all 2048 Y points
// with V_WMMA_F32_16X16X4_F32.
//
// Since this kernel only takes mins over N and ||x_M||^2 is constant in N,
// the scan computes D[M][N] = ||y_N||^2 - 2*x_M.y_N via:
//   A (16x4): K=0..2 -> -2*x,-2*y,-2*z ; K=3 -> 1.0
//   B (4x16): K=0..2 ->  y.x,y.y,y.z   ; K=3 -> ||y_N||^2
//   C = inline 0
// ||x_M||^2 is added once after the min-reduction.
//
// The per-lane B operand pair is prepacked in LDS:
//   lo lanes (K=0,1): {y.x, y.y}   hi lanes (K=2,3): {y.z, ||y||^2}
// so the inner loop is 1 ds_load_b64 + 1 wmma + mins.
__global__ __launch_bounds__(256) void chamfer_rowmin_kernel(
    const float* __restrict__ X, const float* __restrict__ Y,
    float* __restrict__ rowMinOut)
{
  __shared__ __align__(16) float ldsStage[NPTS * 3];      // 24 KB: AoS staging
  __shared__ __align__(16) float ldsB[2 * NPTS + 32 + 2 * NPTS];  // packed B pairs

  const int t = threadIdx.x;
  const int b = blockIdx.y;
  const float* Yb = Y + (size_t)b * NPTS * 3;

  // ---- Async stage 24 KB of Y directly into LDS (GLOBAL_LOAD_ASYNC_TO_LDS_B128,
  // ASYNCcnt-tracked; 1536 16-byte chunks, 6 per thread). ----
  {
    const uint32_t ldsBase = (uint32_t)(uintptr_t)&ldsStage[0];
    const uint64_t gBase   = (uint64_t)(uintptr_t)Yb;
#pragma unroll
    for (int k = 0; k < 6; ++k) {
      const uint32_t c = (uint32_t)t + (uint32_t)k * 256u;  // chunk id
      const uint32_t l = ldsBase + c * 16u;
      const uint64_t g = gBase + (uint64_t)c * 16u;
      asm volatile("global_load_async_to_lds_b128 %0, %1, off"
                   :: "v"(l), "v"(g) : "memory");
    }
    asm volatile("s_wait_asynccnt 0x0" ::: "memory");
  }
  __syncthreads();

  // Pack per-lane B operand pairs.
  for (int i = t; i < NPTS; i += 256) {
    float yx = ldsStage[3 * i + 0], yy = ldsStage[3 * i + 1], yz = ldsStage[3 * i + 2];
    v2f lo; lo.x = yx; lo.y = yy;
    v2f hv; hv.x = yz; hv.y = yx * yx + yy * yy + yz * yz;
    *(v2f*)&ldsB[2 * i]         = lo;
    *(v2f*)&ldsB[HIOFF + 2 * i] = hv;
  }
  __syncthreads();

  const int wave = t >> 5;
  const int lane = t & 31;
  const int hi   = lane >> 4;    // 0 = lanes 0-15 (K=0,1), 1 = lanes 16-31 (K=2,3)
  const int lr   = lane & 15;
  const int rowBase = blockIdx.x * 128 + wave * 16;

  // This lane's X row (row rowBase+lr for both halves of the wave).
  const float* xp = X + ((size_t)b * NPTS + rowBase + lr) * 3;
  const float xx = xp[0], xy = xp[1], xz = xp[2];
  const float x2own = xx * xx + xy * xy + xz * xz;

  // A operand: K=0..2 carry -2*coords, K=3 carries 1.0.
  v2f a;
  a.x = hi ? (-2.0f * xz) : (-2.0f * xx);  // VGPR0: K=2 : K=0
  a.y = hi ? 1.0f : (-2.0f * xy);          // VGPR1: K=3 : K=1

  // Per-lane B base: lo/hi region selected once, plus this lane's column residue.
  const float* bbase = ldsB + (hi ? HIOFF : 0) + 2 * lr;

  float rowMin[8];
#pragma unroll
  for (int r = 0; r < 8; ++r) rowMin[r] = FLT_MAX;

  for (int j = 0; j < NPTS / 16; ++j) {
    const v2f bv = *(const v2f*)(bbase + 32 * j);  // one ds_load_b64

    v8f c = {};                                    // SRC2 = inline 0
    c = __builtin_amdgcn_wmma_f32_16x16x4_f32(
        /*neg_a=*/false, a, /*neg_b=*/false, bv,
        /*c_mod=*/(short)0, c, /*reuse_a=*/false, /*reuse_b=*/false);

#pragma unroll
    for (int r = 0; r < 8; ++r)
      rowMin[r] = fminf(rowMin[r], c[r]);  // c[r] = ||y_N||^2 - 2*x_M.y_N
  }

  // Fold mins across the 16 lanes of each half-wave (they cover all N residues),
  // then add this row's ||x_M||^2 (M = r + 8*hi).
#pragma unroll
  for (int r = 0; r < 8; ++r) {
#pragma unroll
    for (int m = 1; m <= 8; m <<= 1)
      rowMin[r] = fminf(rowMin[r], __shfl_xor(rowMin[r], m, 32));
    rowMin[r] += __shfl(x2own, r + (hi << 3), 32);
  }

  if (lr == 0) {
    float* o = rowMinOut + (size_t)b * NPTS + rowBase + (hi << 3);
#pragma unroll
    for (int r = 0; r < 8; ++r) o[r] = rowMin[r];
  }
}

// Single-block deterministic reduction: mean of dist1+dist2 plus NLL term.
__global__ __launch_bounds__(256) void finalize_kernel(
    const float* __restrict__ dist1, const float* __restrict__ dist2,
    const float* __restrict__ pred, const int* __restrict__ target,
    float* __restrict__ out)
{
  __shared__ float red[256];
  const int t = threadIdx.x;
  float acc = 0.0f;
  for (int i = t; i < BB * NPTS; i += 256)
    acc += dist1[i] + dist2[i];
  red[t] = acc;
  __syncthreads();
  for (int s = 128; s > 0; s >>= 1) {
    if (t < s) red[t] += red[t + s];
    __syncthreads();
  }
  if (t == 0) {
    float nll = 0.0f;
    for (int bidx = 0; bidx < BB; ++bidx)
      nll += pred[bidx * NCLS + target[bidx]];
    nll = -nll / (float)BB;
    out[0] = nll + red[0] / (float)(BB * NPTS);
  }
}

extern "C" void kernel_launch(void* const* d_in, const int* in_sizes, int n_in,
                              void* d_out, int out_size, void* d_ws, size_t ws_size,
                              hipStream_t stream) {
  const float* reg    = (const float*)d_in[0];  // [16,2048,3] f32
  const float* point1 = (const float*)d_in[1];  // [16,2048,3] f32
  const float* pred   = (const float*)d_in[2];  // [16,40]     f32 (log-probs)
  const int*   target = (const int*)d_in[3];    // [16]        i32
  float* out = (float*)d_out;                   // scalar f32

  float* dist1 = (float*)d_ws;                  // [16*2048] f32
  float* dist2 = dist1 + BB * NPTS;             // [16*2048] f32

  dim3 grid(NPTS / 128, BB);                    // (16 row-tiles, 16 batches)
  dim3 block(256);

  chamfer_rowmin_kernel<<<grid, block, 0, stream>>>(reg, point1, dist1);
  chamfer_rowmin_kernel<<<grid, block, 0, stream>>>(point1, reg, dist2);
  finalize_kernel<<<1, block, 0, stream>>>(dist1, dist2, pred, target, out);
}